// EagleWrapper_41996190221113
// MI455X (gfx1250) — compile-verified
//
#include <hip/hip_runtime.h>
#include <hip/hip_bf16.h>
#include <stdint.h>

// Problem shape (fixed by the reference)
#define B_      8
#define S_      1024
#define V_      32000

// Argmax kernel geometry: 320 threads (10 wave32) * float4 = 1280 floats/tile,
// 25 tiles == 32000 exactly.
#define BLOCK_A 320
#define TILE_F  1280
#define NTILE   25
#define DEPTH   4          // async pipeline depth (tiles in flight per wave)

#define USE_ASYNC_LDS 1    // CDNA5 async global->LDS streaming path

// ---------------------------------------------------------------------------
// CDNA5 split-counter waits (builtin if the toolchain has it, else inline asm)
// ---------------------------------------------------------------------------
template <int N> __device__ __forceinline__ void wait_asynccnt() {
#if __has_builtin(__builtin_amdgcn_s_wait_asynccnt)
    __builtin_amdgcn_s_wait_asynccnt(N);
#else
    asm volatile("s_wait_asynccnt %0" :: "n"(N) : "memory");
#endif
}

template <int N> __device__ __forceinline__ void wait_dscnt() {
#if __has_builtin(__builtin_amdgcn_s_wait_dscnt)
    __builtin_amdgcn_s_wait_dscnt(N);
#else
    asm volatile("s_wait_dscnt %0" :: "n"(N) : "memory");
#endif
}

// Per-lane async copy of 16B global -> LDS (GLOBAL_LOAD_ASYNC_TO_LDS_B128,
// tracked by ASYNCcnt). LDS operand is the wave-relative LDS byte offset in a
// VGPR; truncating the generic address of a __shared__ object yields exactly
// that (flat-aperture LDS addresses carry the offset in addr[31:0]).
__device__ __forceinline__ void async_load_b128(const float* gsrc, float* ldst) {
#if USE_ASYNC_LDS
    uint32_t lds_off = (uint32_t)(uintptr_t)ldst;
    asm volatile("global_load_async_to_lds_b128 %0, %1, off"
                 :: "v"(lds_off), "v"(gsrc)
                 : "memory");
#else
    (void)gsrc; (void)ldst;
#endif
}

__device__ __forceinline__ void upd(float& bv, int& bi, float v, int i) {
    // strict > keeps the FIRST occurrence of the max within each thread's
    // ascending-index scan (matches jnp.argmax tie-breaking)
    if (v > bv) { bv = v; bi = i; }
}

__device__ __forceinline__ void merge(float& bv, int& bi, float ov, int oi) {
    // lexicographic (max value, min index)
    if (ov > bv || (ov == bv && oi < bi)) { bv = ov; bi = oi; }
}

// ---------------------------------------------------------------------------
// Kernel A: row-wise argmax over V=32000. One 320-thread WG per (b,s) row.
// HBM-bound: streams 1.05 GB once => ~45 us floor at 23.3 TB/s.
// ---------------------------------------------------------------------------
__global__ __launch_bounds__(BLOCK_A)
void eagle_argmax_kernel(const float* __restrict__ logits, int* __restrict__ greedy) {
    const int row = blockIdx.x;            // 0 .. B_*S_-1
    const int tid = threadIdx.x;           // 0 .. 319
    const float* rowp = logits + (size_t)row * V_;

    float bestV = -3.402823466e+38f;
    int   bestI = 0;

#if USE_ASYNC_LDS
    __shared__ __align__(16) float buf[DEPTH][TILE_F];   // 20 KB double^2 buffer

    // Prologue: fill the pipeline (DEPTH async tile loads in flight per wave).
    #pragma unroll
    for (int d = 0; d < DEPTH; ++d)
        async_load_b128(rowp + d * TILE_F + tid * 4, &buf[d][tid * 4]);

    for (int it = 0; it < NTILE; ++it) {
        // Oldest outstanding async load is tile `it`; wait for it.
        if (it <= NTILE - DEPTH) wait_asynccnt<DEPTH - 1>();
        else                     wait_asynccnt<0>();        // drain at the tail

        const int bsel = it & (DEPTH - 1);
        const float4 v = *(const float4*)&buf[bsel][tid * 4];  // ds_load_b128 (own slot)
        const int base = it * TILE_F + tid * 4;
        upd(bestV, bestI, v.x, base + 0);
        upd(bestV, bestI, v.y, base + 1);
        upd(bestV, bestI, v.z, base + 2);
        upd(bestV, bestI, v.w, base + 3);

        const int nx = it + DEPTH;
        if (nx < NTILE) {
            // WAR guard: ds read of this slot must retire before the async
            // engine overwrites it.
            wait_dscnt<0>();
            async_load_b128(rowp + nx * TILE_F + tid * 4, &buf[bsel][tid * 4]);
        }
    }
#else
    for (int it = 0; it < NTILE; ++it) {
        const float4 v = *(const float4*)(rowp + it * TILE_F + tid * 4);
        const int base = it * TILE_F + tid * 4;
        upd(bestV, bestI, v.x, base + 0);
        upd(bestV, bestI, v.y, base + 1);
        upd(bestV, bestI, v.z, base + 2);
        upd(bestV, bestI, v.w, base + 3);
    }
#endif

    // wave32 shuffle reduction (5 steps)
    #pragma unroll
    for (int off = 16; off > 0; off >>= 1) {
        float ov = __shfl_xor(bestV, off, 32);
        int   oi = __shfl_xor(bestI, off, 32);
        merge(bestV, bestI, ov, oi);
    }

    __shared__ float sV[BLOCK_A / 32];
    __shared__ int   sI[BLOCK_A / 32];
    if ((tid & 31) == 0) { sV[tid >> 5] = bestV; sI[tid >> 5] = bestI; }
    __syncthreads();
    if (tid == 0) {
        float bv = sV[0]; int bi = sI[0];
        #pragma unroll
        for (int w = 1; w < BLOCK_A / 32; ++w) merge(bv, bi, sV[w], sI[w]);
        greedy[row] = bi;
    }
}

// ---------------------------------------------------------------------------
// Kernel B: per-batch verify scan + draft row write. One 1024-thread WG per b.
// Wave 0 does a ballot-based first-mismatch search (32 positions / step).
// ---------------------------------------------------------------------------
__global__ __launch_bounds__(1024)
void eagle_verify_kernel(const int* __restrict__ ids,
                         const int* __restrict__ npa,
                         const int* __restrict__ greedy,
                         float* __restrict__ out,
                         int* __restrict__ bonus_pos_ws) {
    const int b   = blockIdx.x;
    const int tid = threadIdx.x;

    __shared__ int s_num_acc, s_bonus_tok;

    if (tid < 32) {                       // exactly one wave32
        const int start = npa[b] - 1;     // first unchecked logit index
        const int limit = S_ - 1;
        int first = limit;                // "no mismatch" sentinel
        for (int base = start; base < limit; base += 32) {
            const int k = base + tid;
            const bool mm = (k < limit) && (greedy[b * S_ + k] != ids[b * S_ + k + 1]);
            const unsigned long long mask = __ballot(mm);   // wave32: low 32 bits
            if (mask) { first = base + (__ffsll((long long)mask) - 1); break; }
        }
        if (tid == 0) {
            const int num_newly = first - start;
            const int num_acc   = npa[b] + num_newly;
            const int bpos      = start + num_newly;        // npa-1+num_newly
            s_num_acc   = num_acc;
            s_bonus_tok = greedy[b * S_ + bpos];
            bonus_pos_ws[b] = bpos;
            out[B_ * S_ + b]      = (float)num_newly;       // output 2
            out[B_ * S_ + B_ + b] = (float)num_acc;         // output 3
        }
    }
    __syncthreads();

    const int num_acc = s_num_acc;
    int val;
    if      (tid <  num_acc - 1) val = ids[b * S_ + tid + 1];
    else if (tid == num_acc - 1) val = s_bonus_tok;
    else                         val = 0;
    out[b * S_ + tid] = (float)val;                         // output 1 (draft)
}

// ---------------------------------------------------------------------------
// Kernel C: gather last_logits[b] = target_logits[b, bonus_pos[b], :].
// grid (25, 8) x 320 threads, one float4 per thread (8000 float4 per row).
// ---------------------------------------------------------------------------
__global__ __launch_bounds__(BLOCK_A)
void eagle_gather_logits_kernel(const float* __restrict__ logits,
                                const int* __restrict__ bonus_pos_ws,
                                float* __restrict__ out) {
    const int b   = blockIdx.y;
    const int pos = bonus_pos_ws[b];
    const int q   = blockIdx.x * BLOCK_A + threadIdx.x;     // 0..7999
    const float4 v = *(const float4*)(logits + ((size_t)b * S_ + pos) * V_ + (size_t)q * 4);
    float* dst = out + (size_t)B_ * S_ + 2 * B_ + (size_t)b * V_ + (size_t)q * 4;
    *(float4*)dst = v;                                      // output 4
}

// ---------------------------------------------------------------------------
extern "C" void kernel_launch(void* const* d_in, const int* in_sizes, int n_in,
                              void* d_out, int out_size, void* d_ws, size_t ws_size,
                              hipStream_t stream) {
    (void)in_sizes; (void)n_in; (void)out_size; (void)ws_size;
    const int*   input_ids = (const int*)  d_in[0];   // [B,S]
    const float* logits    = (const float*)d_in[1];   // [B,S,V]
    const int*   npa       = (const int*)  d_in[2];   // [B]
    float*       out       = (float*)d_out;

    int* greedy    = (int*)d_ws;                      // [B*S]
    int* bonus_pos = greedy + B_ * S_;                // [B]

    eagle_argmax_kernel<<<dim3(B_ * S_), dim3(BLOCK_A), 0, stream>>>(logits, greedy);
    eagle_verify_kernel<<<dim3(B_), dim3(1024), 0, stream>>>(input_ids, npa, greedy,
                                                             out, bonus_pos);
    eagle_gather_logits_kernel<<<dim3(NTILE, B_), dim3(BLOCK_A), 0, stream>>>(logits,
                                                                              bonus_pos, out);
}